// _LayerPagedCache_8108898255480
// MI455X (gfx1250) — compile-verified
//
#include <hip/hip_runtime.h>
#include <cstdint>
#include <cstddef>

// Problem geometry (from the reference file)
#define PAGE_SIZE     128
#define N_PAGES       256
#define N_KV_HEADS    8
#define HEAD_DIM      128
#define PAGES_PER_SEQ 64
#define SLOT_ELEMS    (N_KV_HEADS * HEAD_DIM)   // 1024 floats = 4 KiB per (page, offset) slot
#define N_SLOTS       (N_PAGES * PAGE_SIZE)     // 32768 slots per cache

// CDNA5 async global<->LDS DMA path (gfx1250-only; guarded so host pass and
// older toolchains still compile via the nontemporal fallback).
#if defined(__gfx1250__) && \
    __has_builtin(__builtin_amdgcn_global_load_async_to_lds_b128) && \
    __has_builtin(__builtin_amdgcn_global_store_async_from_lds_b128)
#define HAVE_ASYNC_LDS 1
#else
#define HAVE_ASYNC_LDS 0
#endif

typedef float v4f __attribute__((ext_vector_type(4)));
typedef int   v4i __attribute__((ext_vector_type(4)));

#if HAVE_ASYNC_LDS
// Clang's gfx1250 async builtins take (AS1 v4i*, AS3 v4i*, imm offset, imm cpol).
typedef __attribute__((address_space(1))) v4i* gv4i_p;
typedef __attribute__((address_space(3))) v4i* lv4i_p;
#endif

__device__ __forceinline__ void wait_asynccnt0() {
#if defined(__gfx1250__)
#if __has_builtin(__builtin_amdgcn_s_wait_asynccnt)
  __builtin_amdgcn_s_wait_asynccnt(0);
#else
  asm volatile("s_wait_asynccnt 0" ::: "memory");
#endif
#endif
}

// ---------------------------------------------------------------------------
// Kernel 1: zero the inverse map (one int per (page, offset) slot).
// ---------------------------------------------------------------------------
__global__ void mark_zero_kernel(int* __restrict__ mark) {
  mark[blockIdx.x * blockDim.x + threadIdx.x] = 0;   // launched with exactly N_SLOTS threads
}

// ---------------------------------------------------------------------------
// Kernel 2: inverse map. mark[phys_page*128 + offset] = token_index + 1.
// pos_ids are unique in the reference; on duplicates any writer winning
// matches jnp scatter's unspecified-order semantics.
// ---------------------------------------------------------------------------
__global__ void mark_set_kernel(const int* __restrict__ page_table,
                                const int* __restrict__ pos_ids,
                                const int* __restrict__ batch_idx,
                                int* __restrict__ mark, int S) {
  int s = blockIdx.x * blockDim.x + threadIdx.x;
  if (s >= S) return;
  int b   = batch_idx[0];                 // low word of the (possibly i64) batch index
  int pos = pos_ids[s];
  int pp  = page_table[b * PAGES_PER_SEQ + (pos >> 7)];
  mark[(pp << 7) | (pos & (PAGE_SIZE - 1))] = s + 1;
}

// ---------------------------------------------------------------------------
// Kernel 3: single-pass gather. One wave per output slot (4 KiB):
//   mark==0 : stream the old cache slot (contiguous 4 KiB)
//   mark==t : gather token t-1 from {k,v}_val (8 x 512 B, one chunk per head)
// All traffic goes through the CDNA5 async DMA path (ASYNCcnt): 8 async
// b128 loads into this wave's LDS tile, s_wait_asynccnt 0, 8 async b128
// stores to the output. No VGPR data staging.
// ---------------------------------------------------------------------------
__global__ __launch_bounds__(256) void paged_gather_kernel(
    const float* __restrict__ k_cache, const float* __restrict__ v_cache,
    const float* __restrict__ k_val,   const float* __restrict__ v_val,
    const int*   __restrict__ mark,    // may be null => copy everything
    float* __restrict__ out, int S)
{
  __shared__ float lds[8 * SLOT_ELEMS];               // 8 waves x 4 KiB = 32 KiB
  const int wave = threadIdx.x >> 5;
  const int lane = threadIdx.x & 31;
  const int slot = (int)blockIdx.x * 8 + wave;        // [0, 2*N_SLOTS)
  const int c       = slot >> 15;                     // 0 = K cache, 1 = V cache
  const int po_slot = slot & (N_SLOTS - 1);           // page*128 + offset

  const int m = mark ? mark[po_slot] : 0;             // wave-uniform

  const float* base;
  size_t jstride;                                     // element stride between heads
  if (m == 0) {
    base    = (c ? v_cache : k_cache) + ((size_t)po_slot * SLOT_ELEMS);
    jstride = HEAD_DIM;                               // contiguous 4 KiB
  } else {
    base    = (c ? v_val : k_val) + ((size_t)(m - 1) * HEAD_DIM);
    jstride = (size_t)S * HEAD_DIM;                   // per-head 512 B chunks
  }
  base += lane * 4;

  float* wl  = lds + wave * SLOT_ELEMS + lane * 4;
  float* dst = out + ((size_t)slot * SLOT_ELEMS) + lane * 4;

#if HAVE_ASYNC_LDS
  #pragma unroll
  for (int j = 0; j < N_KV_HEADS; ++j) {
    __builtin_amdgcn_global_load_async_to_lds_b128(
        (gv4i_p)(base + j * jstride),
        (lv4i_p)(wl + j * HEAD_DIM),
        0, 0);
  }
  wait_asynccnt0();                                   // per-wave ASYNCcnt: LDS tile ready
  #pragma unroll
  for (int j = 0; j < N_KV_HEADS; ++j) {
    __builtin_amdgcn_global_store_async_from_lds_b128(
        (gv4i_p)(dst + j * HEAD_DIM),
        (lv4i_p)(wl + j * HEAD_DIM),
        0, 0);
  }
  // s_endpgm performs an implicit wait-idle; no explicit drain needed.
#else
  (void)lds; (void)wl;
  #pragma unroll
  for (int j = 0; j < N_KV_HEADS; ++j) {
    v4f t = __builtin_nontemporal_load((const v4f*)(base + j * jstride));
    __builtin_nontemporal_store(t, (v4f*)(dst + j * HEAD_DIM));
  }
#endif
}

// ---------------------------------------------------------------------------
// Fallback (only if scratch is too small for the mark array):
// direct token scatter after a full cache copy. One wave per (cache, token).
// ---------------------------------------------------------------------------
__global__ __launch_bounds__(256) void scatter_tokens_kernel(
    const float* __restrict__ k_val, const float* __restrict__ v_val,
    const int* __restrict__ page_table, const int* __restrict__ pos_ids,
    const int* __restrict__ batch_idx, float* __restrict__ out, int S)
{
  const int wave = threadIdx.x >> 5;
  const int lane = threadIdx.x & 31;
  const int idx  = (int)blockIdx.x * 8 + wave;        // [0, 2*S)
  if (idx >= 2 * S) return;
  const int c = (idx < S) ? 0 : 1;
  const int s = idx - c * S;

  int b   = batch_idx[0];
  int pos = pos_ids[s];
  int pp  = page_table[b * PAGES_PER_SEQ + (pos >> 7)];
  size_t dstBase = ((size_t)c * N_SLOTS + ((size_t)pp << 7) + (pos & (PAGE_SIZE - 1)))
                   * SLOT_ELEMS;
  const float* src = (c ? v_val : k_val) + ((size_t)s * HEAD_DIM) + lane * 4;
  float* dst = out + dstBase + lane * 4;

  #pragma unroll
  for (int j = 0; j < N_KV_HEADS; ++j) {
    v4f t = *(const v4f*)(src + (size_t)j * S * HEAD_DIM);
    __builtin_nontemporal_store(t, (v4f*)(dst + j * HEAD_DIM));
  }
}

// ---------------------------------------------------------------------------
extern "C" void kernel_launch(void* const* d_in, const int* in_sizes, int n_in,
                              void* d_out, int out_size, void* d_ws, size_t ws_size,
                              hipStream_t stream) {
  const float* k_cache    = (const float*)d_in[0];
  const float* v_cache    = (const float*)d_in[1];
  const float* k_val      = (const float*)d_in[2];
  const float* v_val      = (const float*)d_in[3];
  const int*   page_table = (const int*)  d_in[4];
  const int*   pos_ids    = (const int*)  d_in[5];
  const int*   batch_idx  = (const int*)  d_in[6];
  float*       out        = (float*)d_out;
  const int S = in_sizes[5];                          // number of tokens (8192)

  const int gatherBlocks = (2 * N_SLOTS) / 8;         // 8 waves/block, 1 slot/wave

  if (ws_size >= (size_t)N_SLOTS * sizeof(int)) {
    // Masked single-pass path: 512 MiB total traffic (~22 us @ 23.3 TB/s).
    int* mark = (int*)d_ws;
    mark_zero_kernel<<<N_SLOTS / 256, 256, 0, stream>>>(mark);
    mark_set_kernel<<<(S + 255) / 256, 256, 0, stream>>>(page_table, pos_ids,
                                                         batch_idx, mark, S);
    paged_gather_kernel<<<gatherBlocks, 256, 0, stream>>>(
        k_cache, v_cache, k_val, v_val, mark, out, S);
  } else {
    // Fallback: full copy then scatter (640 MiB traffic).
    paged_gather_kernel<<<gatherBlocks, 256, 0, stream>>>(
        k_cache, v_cache, k_val, v_val, nullptr, out, S);
    scatter_tokens_kernel<<<(2 * S + 7) / 8, 256, 0, stream>>>(
        k_val, v_val, page_table, pos_ids, batch_idx, out, S);
  }
}